// TensorProductScoreModel_60103772340560
// MI455X (gfx1250) — compile-verified
//
#include <hip/hip_runtime.h>

#define NS 32
#define NV 8
#define NNODES 8000
#define NEDGES 160000
#define EF 96
#define HID 96
#define NC 56           // NS + 3*NV
#define WNUM 1600

typedef __attribute__((ext_vector_type(2))) float v2f;
typedef __attribute__((ext_vector_type(4))) float v4f;
typedef __attribute__((ext_vector_type(8))) float v8f;

__device__ __forceinline__ v8f wmma4(v2f a, v2f b, v8f c) {
  // D(16x16,f32) += A(16x4,f32) * B(4x16,f32)
  return __builtin_amdgcn_wmma_f32_16x16x4_f32(false, a, false, b, (short)0, c,
                                               false, false);
}

__global__ __launch_bounds__(256) void zero_ws_kernel(float* ws, int n) {
  int i = blockIdx.x * blockDim.x + threadIdx.x;
  int stride = gridDim.x * blockDim.x;
  for (; i < n; i += stride) ws[i] = 0.0f;
}

__global__ __launch_bounds__(256) void edge_kernel(
    const float* __restrict__ node_attr,   // [8000][56]
    const float* __restrict__ edge_attr,   // [160000][96]
    const float* __restrict__ edge_sh,     // [160000][4]
    const float* __restrict__ W1,          // [96][96]
    const float* __restrict__ b1,          // [96]
    const float* __restrict__ W2,          // [96][1600]
    const float* __restrict__ b2,          // [1600]
    const int*   __restrict__ edge_index,  // [2][160000]
    float* __restrict__ agg,               // [8000][56] accumulators
    float* __restrict__ cnt)               // [8000]
{
  __shared__ float w1t[96][98];            // w1t[n][k] = W1[k][n]
  __shared__ float b1s[96];
  __shared__ float b2s[WNUM];
  __shared__ float w2t[2][16][98];         // double-buffered: w2t[b][c][k] = W2[k][c0+c]
  __shared__ float hBuf[8][16][98];        // per-wave h (16 edges x 96)
  __shared__ float sBuf[8][16][32];        // scalar inputs s
  __shared__ float vBuf[8][16][24];        // vector inputs v (NV x 3)
  __shared__ float bBuf[8][16][8];         // b[j] = sum_d v[j][d]*y1[d]
  __shared__ float y0B[8][16];
  __shared__ float y1B[8][16][3];
  __shared__ int   dstB[8][16];
  __shared__ float tC[8][16][8];           // t[o] = sum_i s_i wC[i][o]
  __shared__ float vAcc[8][16][8][3];      // sum_i y0*v[i][d]*wD[i][o]

  const int tid  = threadIdx.x;
  const int w    = tid >> 5;
  const int lane = tid & 31;
  const int l15  = lane & 15;
  const int hi   = lane >> 4;              // 0: K=0..1 half, 1: K=2..3 half
  const int e0   = blockIdx.x * 128 + w * 16;

  // ---- stage W1 (transposed) + biases, block-wide ----
  for (int idx = tid; idx < 96 * 96; idx += 256) {
    int k = idx / 96, n = idx % 96;
    w1t[n][k] = W1[idx];
  }
  if (tid < 96) b1s[tid] = b1[tid];
  for (int idx = tid; idx < WNUM; idx += 256) b2s[idx] = b2[idx];

  // ---- per-wave edge staging ----
  {
    const int e   = e0 + l15;
    const int src = edge_index[e];
    const v4f* srow = (const v4f*)(node_attr + (size_t)src * NC);
#pragma unroll
    for (int u = 0; u < 4; ++u) {          // s: floats [0..32)
      v4f q = srow[hi * 4 + u];
      sBuf[w][l15][hi * 16 + 4 * u + 0] = q.x;
      sBuf[w][l15][hi * 16 + 4 * u + 1] = q.y;
      sBuf[w][l15][hi * 16 + 4 * u + 2] = q.z;
      sBuf[w][l15][hi * 16 + 4 * u + 3] = q.w;
    }
    if (lane < 16) {
      dstB[w][lane] = edge_index[NEDGES + e];
      v4f q = *(const v4f*)(edge_sh + (size_t)e * 4);
      y0B[w][lane] = q.x;
      y1B[w][lane][0] = q.y; y1B[w][lane][1] = q.z; y1B[w][lane][2] = q.w;
#pragma unroll
      for (int u = 0; u < 6; ++u) {        // v: floats [32..56)
        v4f t = srow[8 + u];
        vBuf[w][lane][4 * u + 0] = t.x;
        vBuf[w][lane][4 * u + 1] = t.y;
        vBuf[w][lane][4 * u + 2] = t.z;
        vBuf[w][lane][4 * u + 3] = t.w;
      }
    }
  }
  // zero per-wave LDS accumulators
  {
    float* tc = &tC[w][0][0];
#pragma unroll
    for (int q = 0; q < 4; ++q) tc[lane + 32 * q] = 0.0f;
    float* va = &vAcc[w][0][0][0];
#pragma unroll
    for (int q = 0; q < 12; ++q) va[lane + 32 * q] = 0.0f;
  }

  // ---- issue async global->LDS copy of W2 chunk 0 (overlaps with GEMM1) ----
  {
    const int kk = tid >> 4, cc = tid & 15;
#pragma unroll
    for (int r = 0; r < 6; ++r) {
      const float* g = W2 + (size_t)(kk + 16 * r) * WNUM + cc;
      unsigned lds = (unsigned)(uintptr_t)&w2t[0][cc][kk + 16 * r];
      asm volatile("global_load_async_to_lds_b32 %0, %1, off"
                   :: "v"(lds), "v"(g) : "memory");
    }
  }

  __syncthreads();

  // ---- bBuf[e][j] = sum_d v[e][j][d] * y1[e][d] ----
#pragma unroll
  for (int q = 0; q < 4; ++q) {
    int slot = lane + 32 * q;
    int e = slot >> 3, j = slot & 7;
    float acc = 0.0f;
#pragma unroll
    for (int d = 0; d < 3; ++d) acc += vBuf[w][e][j * 3 + d] * y1B[w][e][d];
    bBuf[w][e][j] = acc;
  }

  // ---- GEMM1: h = relu(edge_attr(16x96) @ W1(96x96) + b1) ----
  v8f acc1[6];
#pragma unroll
  for (int nt = 0; nt < 6; ++nt)
    acc1[nt] = (v8f){0.f, 0.f, 0.f, 0.f, 0.f, 0.f, 0.f, 0.f};

  const float* Abase = edge_attr + (size_t)(e0 + l15) * EF + 2 * hi;
  for (int ks = 0; ks < 24; ++ks) {
    v2f a = *(const v2f*)(Abase + 4 * ks);
#pragma unroll
    for (int nt = 0; nt < 6; ++nt) {
      v2f b = *(const v2f*)&w1t[nt * 16 + l15][4 * ks + 2 * hi];
      acc1[nt] = wmma4(a, b, acc1[nt]);
    }
  }
#pragma unroll
  for (int nt = 0; nt < 6; ++nt) {
#pragma unroll
    for (int v = 0; v < 8; ++v) {
      float h = acc1[nt][v] + b1s[nt * 16 + l15];
      hBuf[w][v + 8 * hi][nt * 16 + l15] = fmaxf(h, 0.0f);
    }
  }

  // ---- read h back in A-operand layout, keep in registers for all chunks ----
  v2f aReg[24];
#pragma unroll
  for (int ks = 0; ks < 24; ++ks)
    aReg[ks] = *(const v2f*)&hBuf[w][l15][4 * ks + 2 * hi];

  // ---- GEMM2 chunks + fused tensor-product consumption ----
  float outS0[8], outS1[8];
#pragma unroll
  for (int v = 0; v < 8; ++v) { outS0[v] = 0.0f; outS1[v] = 0.0f; }

  const int kk = tid >> 4, cc = tid & 15;
  for (int ch = 0; ch < 100; ++ch) {
    const int cur = ch & 1;
    // prefetch next chunk into the other buffer (async, overlapped)
    if (ch + 1 < 100) {
      const int c0n = (ch + 1) * 16;
#pragma unroll
      for (int r = 0; r < 6; ++r) {
        const float* g = W2 + (size_t)(kk + 16 * r) * WNUM + c0n + cc;
        unsigned lds = (unsigned)(uintptr_t)&w2t[cur ^ 1][cc][kk + 16 * r];
        asm volatile("global_load_async_to_lds_b32 %0, %1, off"
                     :: "v"(lds), "v"(g) : "memory");
      }
      asm volatile("s_wait_asynccnt 0x6" ::: "memory");  // current chunk landed
    } else {
      asm volatile("s_wait_asynccnt 0x0" ::: "memory");
    }
    __syncthreads();                       // all waves' copies of chunk ch visible

    const int c0 = ch * 16;
    v8f acc = (v8f){0.f, 0.f, 0.f, 0.f, 0.f, 0.f, 0.f, 0.f};
#pragma unroll
    for (int ks = 0; ks < 24; ++ks) {
      v2f b = *(const v2f*)&w2t[cur][l15][4 * ks + 2 * hi];
      acc = wmma4(aReg[ks], b, acc);
    }
    const float b2v = b2s[c0 + l15];       // fold bias into w

    if (ch < 64) {                         // wA[i][o]: out_s[o] += s_i*y0 * w
      const int i = ch >> 1, half = ch & 1;
#pragma unroll
      for (int v = 0; v < 8; ++v) {
        const int m = v + 8 * hi;
        const float wv = acc[v] + b2v;
        const float a = sBuf[w][m][i] * y0B[w][m];
        if (half) outS1[v] += a * wv; else outS0[v] += a * wv;
      }
    } else if (ch < 80) {                  // wB[j][o]: out_s[o] += b_j * w
      const int j = (ch - 64) >> 1, half = (ch - 64) & 1;
#pragma unroll
      for (int v = 0; v < 8; ++v) {
        const int m = v + 8 * hi;
        const float wv = acc[v] + b2v;
        const float bb = bBuf[w][m][j];
        if (half) outS1[v] += bb * wv; else outS0[v] += bb * wv;
      }
    } else if (ch < 96) {                  // wC[i][o]: t[o] += s_i * w
      const int i = 2 * (ch - 80) + (l15 >> 3);
      const int o = l15 & 7;
#pragma unroll
      for (int v = 0; v < 8; ++v) {
        const int m = v + 8 * hi;
        const float wv = acc[v] + b2v;
        atomicAdd(&tC[w][m][o], sBuf[w][m][i] * wv);
      }
    } else {                               // wD[i][o]: vAcc[o][d] += y0*v[i][d]*w
      const int i = 2 * (ch - 96) + (l15 >> 3);
      const int o = l15 & 7;
#pragma unroll
      for (int v = 0; v < 8; ++v) {
        const int m = v + 8 * hi;
        const float wv = acc[v] + b2v;
        const float y0v = y0B[w][m];
#pragma unroll
        for (int d = 0; d < 3; ++d)
          atomicAdd(&vAcc[w][m][o][d], y0v * vBuf[w][m][i * 3 + d] * wv);
      }
    }
    __syncthreads();                       // chunk consumed; buffer may be reused
  }

  // ---- emit: scatter-add msg into agg[dst] ----
#pragma unroll
  for (int v = 0; v < 8; ++v) {
    const int e = v + 8 * hi;
    const int base = dstB[w][e] * NC;
    unsafeAtomicAdd(&agg[base + l15], outS0[v]);
    unsafeAtomicAdd(&agg[base + 16 + l15], outS1[v]);
  }
#pragma unroll
  for (int q = 0; q < 4; ++q) {
    const int slot = lane + 32 * q;
    const int e = slot >> 3, o = slot & 7;
    const int base = dstB[w][e] * NC + NS + o * 3;
    const float t = tC[w][e][o];
#pragma unroll
    for (int d = 0; d < 3; ++d)
      unsafeAtomicAdd(&agg[base + d], t * y1B[w][e][d] + vAcc[w][e][o][d]);
  }
  if (lane < 16) unsafeAtomicAdd(&cnt[dstB[w][lane]], 1.0f);
}

__global__ __launch_bounds__(256) void finalize_kernel(
    const float* __restrict__ agg, const float* __restrict__ cnt,
    const float* __restrict__ node_attr, float* __restrict__ out) {
  int idx = blockIdx.x * blockDim.x + threadIdx.x;
  if (idx >= NNODES * NC) return;
  int n = idx / NC;
  out[idx] = agg[idx] / fmaxf(cnt[n], 1.0f) + node_attr[idx];
}

extern "C" void kernel_launch(void* const* d_in, const int* in_sizes, int n_in,
                              void* d_out, int out_size, void* d_ws, size_t ws_size,
                              hipStream_t stream) {
  const float* node_attr = (const float*)d_in[0];
  const float* edge_attr = (const float*)d_in[1];
  const float* edge_sh   = (const float*)d_in[2];
  const float* W1        = (const float*)d_in[3];
  const float* b1        = (const float*)d_in[4];
  const float* W2        = (const float*)d_in[5];
  const float* b2        = (const float*)d_in[6];
  const int*   edge_index = (const int*)d_in[7];
  float* out = (float*)d_out;

  float* agg = (float*)d_ws;               // 8000*56 floats
  float* cnt = agg + NNODES * NC;          // 8000 floats

  zero_ws_kernel<<<512, 256, 0, stream>>>(agg, NNODES * NC + NNODES);
  edge_kernel<<<NEDGES / 128, 256, 0, stream>>>(
      node_attr, edge_attr, edge_sh, W1, b1, W2, b2, edge_index, agg, cnt);
  finalize_kernel<<<(NNODES * NC + 255) / 256, 256, 0, stream>>>(
      agg, cnt, node_attr, out);
}